// Capsule_56169582297695
// MI455X (gfx1250) — compile-verified
//
#include <hip/hip_runtime.h>
#include <hip/hip_bf16.h>

// Problem dims (fixed by reference)
#define Bb 32
#define Nn 2048
#define Kk 32
#define Dd 16
#define NC 16                 // n-chunks for caps GEMM
#define CHUNK (Nn / NC)       // 128
#define TB 16                 // n-rows per TDM batch (16 * 1KB = 16KB per buffer)
#define NBATCH (CHUNK / TB)   // 8
#define LOG2PI 1.8378770664093453f
#define EPSQ 1e-7f

typedef __attribute__((ext_vector_type(2))) float v2f;
typedef __attribute__((ext_vector_type(8))) float v8f;
typedef __attribute__((ext_vector_type(4))) unsigned int v4u;
typedef __attribute__((ext_vector_type(8))) int v8i;
typedef __attribute__((ext_vector_type(4))) int v4i;

// ---------------------------------------------------------------- TDM issue
// Build a Tensor DMA Descriptor (D#) for a 2D tile of W:
//   dim0: 256 contiguous floats  (= W[n,k,:, :] for one (n,k))
//   dim1: `rows` n-values, stride Kk*256 floats between rows
// and issue TENSOR_LOAD_TO_LDS (tracked by TENSORcnt).
__device__ __forceinline__ void tdm_load_w_tile(const float* gptr, unsigned lds_off,
                                                int rows) {
    unsigned long long ga = (unsigned long long)(__UINTPTR_TYPE__)gptr;
    v4u g0;
    g0.x = 1u;                                   // count=1 (valid), user mode
    g0.y = lds_off;                              // LDS byte address
    g0.z = (unsigned)(ga & 0xffffffffu);         // global_addr[31:0]
    g0.w = (unsigned)((ga >> 32) & 0x1ffffffu)   // global_addr[56:32]
           | (2u << 30);                         // type = 2 ("image")
    v8i g1;
    g1[0] = (int)(2u << 16);                     // workgroup_mask=0, data_size=2 (4B)
    g1[1] = (int)(256u << 16);                   // tensor_dim0[15:0] = 256
    g1[2] = (int)((unsigned)rows << 16);         // tensor_dim0 hi=0 | tensor_dim1 lo
    g1[3] = (int)(256u << 16);                   // tensor_dim1 hi=0 | tile_dim0 = 256
    g1[4] = rows;                                // tile_dim1 = rows, tile_dim2 = 0
    g1[5] = (int)(Kk * 256u);                    // tensor_dim0_stride = 8192 elements
    g1[6] = 0;                                   // stride hi | tensor_dim1_stride lo
    g1[7] = 0;                                   // tensor_dim1_stride hi
    v4i g2 = {0, 0, 0, 0};                       // 2D tensor: dims 2/3 unused
    v4i g3 = {0, 0, 0, 0};
    v8i g4 = {0, 0, 0, 0, 0, 0, 0, 0};           // extra descriptor group (clang-23 form)
    __builtin_amdgcn_tensor_load_to_lds(g0, g1, g2, g3, g4, 0);
}

// ---------------------------------------------------------------- init params
__global__ void cap_init(const float* __restrict__ mu0, float* __restrict__ mu,
                         float* __restrict__ is2, float* __restrict__ coef) {
    int i = blockIdx.x * blockDim.x + threadIdx.x;
    if (i < Bb * Kk * Dd) {
        mu[i]  = mu0[i];
        is2[i] = 1.0f;
    }
    if (i < Bb * Kk) {
        coef[i] = -3.4657359028f - 0.5f * Dd * LOG2PI;   // log(1/32) - 8*log(2pi)
    }
}

// ---------------------------------------------------------------- E-step
// R[b,n,k] = softmax_k( coef[b,k] - 0.5 * sum_d (x-mu)^2 * invsig2 )
__global__ void __launch_bounds__(128) cap_estep(
    const float* __restrict__ x, const float* __restrict__ mu,
    const float* __restrict__ is2, const float* __restrict__ coef,
    float* __restrict__ R) {
    __shared__ float smu[Kk * Dd];
    __shared__ float sis[Kk * Dd];
    __shared__ float scf[Kk];
    int b = blockIdx.x;
    int tid = threadIdx.x;
    for (int i = tid; i < Kk * Dd; i += 128) {
        smu[i] = mu[b * Kk * Dd + i];
        sis[i] = is2[b * Kk * Dd + i];
    }
    if (tid < Kk) scf[tid] = coef[b * Kk + tid];
    __syncthreads();

    int n = blockIdx.y * 128 + tid;
    const float* xp = x + (size_t)(b * Nn + n) * Dd;
    float xv[Dd];
#pragma unroll
    for (int dd = 0; dd < Dd; ++dd) xv[dd] = xp[dd];

    float logits[Kk];
    float mx = -1e30f;
    for (int kc = 0; kc < Kk; ++kc) {
        float acc = 0.0f;
#pragma unroll
        for (int dd = 0; dd < Dd; ++dd) {
            float diff = xv[dd] - smu[kc * Dd + dd];
            acc = fmaf(diff * diff, sis[kc * Dd + dd], acc);
        }
        float lg = scf[kc] - 0.5f * acc;
        logits[kc] = lg;
        mx = fmaxf(mx, lg);
    }
    float sum = 0.0f;
    for (int kc = 0; kc < Kk; ++kc) {
        float e = __expf(logits[kc] - mx);
        logits[kc] = e;
        sum += e;
    }
    float inv = 1.0f / sum;
    float* Rp = R + (size_t)(b * Nn + n) * Kk;
    for (int kc = 0; kc < Kk; ++kc) Rp[kc] = logits[kc] * inv;
}

// ---------------------------------------------------------------- N_k reduce
__global__ void __launch_bounds__(256) cap_nk(const float* __restrict__ R,
                                              float* __restrict__ Nk) {
    __shared__ float sdata[256];
    int bk = blockIdx.x;            // b*32 + k
    int b = bk >> 5, kc = bk & 31;
    float acc = 0.0f;
    for (int n = threadIdx.x; n < Nn; n += 256)
        acc += R[(size_t)(b * Nn + n) * Kk + kc];
    sdata[threadIdx.x] = acc;
    __syncthreads();
    for (int s = 128; s > 0; s >>= 1) {
        if ((int)threadIdx.x < s) sdata[threadIdx.x] += sdata[threadIdx.x + s];
        __syncthreads();
    }
    if (threadIdx.x == 0) Nk[bk] = sdata[0];
}

// ---------------------------------------------------------------- M-step GEMM (WMMA fp32)
// S1[b,k,d] = sum_n R[b,n,k]*x[b,n,d]   (ntile 0)
// S2[b,k,d] = sum_n R[b,n,k]*x[b,n,d]^2 (ntile 1)
__global__ void __launch_bounds__(32) cap_mstep_gemm(
    const float* __restrict__ R, const float* __restrict__ x,
    float* __restrict__ S1, float* __restrict__ S2) {
    int b = blockIdx.x, mt = blockIdx.y, nt = blockIdx.z;
    int lane = threadIdx.x;
    int half = lane >> 4;       // K-pair select: 0 -> K0,K1 ; 1 -> K2,K3
    int l16  = lane & 15;
    int kc   = mt * 16 + l16;   // A-matrix row (capsule)
    int dcol = l16;             // B-matrix col (dim)

    const float* Rb = R + (size_t)b * Nn * Kk;
    const float* xb = x + (size_t)b * Nn * Dd;

    v8f acc = {};
    for (int n0 = 0; n0 < Nn; n0 += 4) {
        int na = n0 + half * 2;
        v2f a, bv;
        a.x = Rb[(size_t)(na + 0) * Kk + kc];
        a.y = Rb[(size_t)(na + 1) * Kk + kc];
        float b0 = xb[(size_t)(na + 0) * Dd + dcol];
        float b1 = xb[(size_t)(na + 1) * Dd + dcol];
        if (nt) { b0 = b0 * b0; b1 = b1 * b1; }
        bv.x = b0; bv.y = b1;
        acc = __builtin_amdgcn_wmma_f32_16x16x4_f32(
            false, a, false, bv, (short)0, acc, false, false);
    }
    float* out = (nt ? S2 : S1) + (size_t)b * Kk * Dd;
#pragma unroll
    for (int r = 0; r < 8; ++r) {
        int row = mt * 16 + r + half * 8;   // C/D layout: M = r (+8 for lanes 16-31)
        out[row * Dd + dcol] = acc[r];
    }
}

// ---------------------------------------------------------------- finalize params
__global__ void cap_finalize(const float* __restrict__ S1, const float* __restrict__ S2,
                             const float* __restrict__ Nk, float* __restrict__ mu,
                             float* __restrict__ is2, float* __restrict__ coef) {
    int bk = blockIdx.x * blockDim.x + threadIdx.x;
    if (bk >= Bb * Kk) return;
    float nk = Nk[bk];
    float invnk = 1.0f / nk;
    float sumlog = 0.0f;
#pragma unroll
    for (int dd = 0; dd < Dd; ++dd) {
        float m  = S1[bk * Dd + dd] * invnk;
        float s2 = S2[bk * Dd + dd] * invnk - m * m;
        s2 = fmaxf(s2, 1e-20f);
        mu[bk * Dd + dd]  = m;
        is2[bk * Dd + dd] = 1.0f / s2;
        sumlog += __logf(s2);
    }
    coef[bk] = __logf(nk * (1.0f / (float)Nn)) - 0.5f * Dd * LOG2PI - 0.5f * sumlog;
}

// ---------------------------------------------------------------- caps GEMM (WMMA fp32 + TDM)
// s[b,k,d] = sum_{n,e} (R[b,n,k]*x[b,n,e]) * W[n,k,d,e]
// One 2-wave block per (k, n-chunk); wave = b-tile. Wave 0 drives the Tensor
// Data Mover: W[n..n+TB, k, :, :] tiles are DMA'd into a double-buffered LDS
// staging area (TENSORcnt), overlapped with the WMMA chain of the previous tile.
__global__ void __launch_bounds__(64) cap_caps_gemm(
    const float* __restrict__ W, const float* __restrict__ x,
    const float* __restrict__ R, float* __restrict__ part) {
    __shared__ float smemW[2][TB * 256];   // assumed at LDS offset 0 (sole LDS object)
    int kc = blockIdx.x, ch = blockIdx.y;
    int wid  = threadIdx.x >> 5;   // wave id = b-tile
    int lane = threadIdx.x & 31;
    int half = lane >> 4;
    int l16  = lane & 15;
    int bidx = wid * 16 + l16;     // A-matrix row (batch)
    int dcol = l16;                // B-matrix col (capsule dim)
    int eoff = half * 2;           // e-pair select within group of 4

    int nstart = ch * CHUNK;
    if (wid == 0) {
        tdm_load_w_tile(W + ((size_t)nstart * Kk + kc) * 256, 0u, TB);
    }

    v8f acc = {};
    for (int ib = 0; ib < NBATCH; ++ib) {
        int nbase = nstart + ib * TB;
        if (wid == 0) {
            if (ib + 1 < NBATCH) {
                tdm_load_w_tile(W + ((size_t)(nbase + TB) * Kk + kc) * 256,
                                (unsigned)(((ib + 1) & 1) * TB * 256 * 4), TB);
                __builtin_amdgcn_s_wait_tensorcnt(1);   // batch ib landed in LDS
            } else {
                __builtin_amdgcn_s_wait_tensorcnt(0);
            }
        }
        __syncthreads();   // publish LDS tile to both waves

        const float* wb = &smemW[ib & 1][0];
        for (int nl = 0; nl < TB; ++nl) {
            int n = nbase + nl;
            float r = R[(size_t)(bidx * Nn + n) * Kk + kc];
            const float* xp = x + (size_t)(bidx * Nn + n) * Dd;
            const float* wp = wb + nl * 256 + dcol * 16;
#pragma unroll
            for (int t = 0; t < 4; ++t) {
                v2f xv = *(const v2f*)(xp + 4 * t + eoff);
                v2f a, bv;
                a.x = r * xv.x;
                a.y = r * xv.y;
                bv = *(const v2f*)(wp + 4 * t + eoff);   // ds_load_b64 from staged tile
                acc = __builtin_amdgcn_wmma_f32_16x16x4_f32(
                    false, a, false, bv, (short)0, acc, false, false);
            }
        }
        __syncthreads();   // tile consumed; safe to overwrite next iteration
    }

    float* p = part + ((size_t)(kc * 2 + wid) * NC + ch) * 256;
#pragma unroll
    for (int r = 0; r < 8; ++r)
        p[(r + half * 8) * 16 + dcol] = acc[r];
}

// ---------------------------------------------------------------- reduce + squash
__global__ void __launch_bounds__(256) cap_reduce_squash(
    const float* __restrict__ part, float* __restrict__ out) {
    __shared__ float sdata[16][16];
    int kc = blockIdx.x, bt = blockIdx.y;
    int tid = threadIdx.x;
    int m = tid >> 4;       // batch row within tile
    int dcol = tid & 15;
    const float* p = part + (size_t)(kc * 2 + bt) * NC * 256;
    float s = 0.0f;
    for (int ch = 0; ch < NC; ++ch) s += p[ch * 256 + m * 16 + dcol];
    sdata[m][dcol] = s * s;
    __syncthreads();
    for (int stride = 8; stride > 0; stride >>= 1) {
        if (dcol < stride) sdata[m][dcol] += sdata[m][dcol + stride];
        __syncthreads();
    }
    float ss = sdata[m][0] + EPSQ;
    float scale = sqrtf(ss) / (1.0f + ss);
    int b = bt * 16 + m;
    out[(size_t)(b * Kk + kc) * Dd + dcol] = scale * s;
}

// ---------------------------------------------------------------- launch
extern "C" void kernel_launch(void* const* d_in, const int* in_sizes, int n_in,
                              void* d_out, int out_size, void* d_ws, size_t ws_size,
                              hipStream_t stream) {
    (void)in_sizes; (void)n_in; (void)out_size; (void)ws_size;
    const float* x   = (const float*)d_in[0];   // [32,2048,16]
    const float* W   = (const float*)d_in[1];   // [1,2048,32,16,16]
    const float* mu0 = (const float*)d_in[2];   // [32,32,16]
    float* out = (float*)d_out;                 // [32,32,16]

    // workspace layout (floats)
    float* ws   = (float*)d_ws;
    float* R    = ws;                         // 32*2048*32 = 2,097,152
    float* mu   = R + (size_t)Bb * Nn * Kk;   // 16384
    float* is2  = mu + Bb * Kk * Dd;          // 16384
    float* coef = is2 + Bb * Kk * Dd;         // 1024
    float* Nk   = coef + Bb * Kk;             // 1024
    float* S1   = Nk + Bb * Kk;               // 16384
    float* S2   = S1 + Bb * Kk * Dd;          // 16384
    float* part = S2 + Bb * Kk * Dd;          // 32*2*16*256 = 262,144

    cap_init<<<(Bb * Kk * Dd + 255) / 256, 256, 0, stream>>>(mu0, mu, is2, coef);

    for (int it = 0; it < 3; ++it) {
        cap_estep<<<dim3(Bb, Nn / 128), 128, 0, stream>>>(x, mu, is2, coef, R);
        if (it < 2) {  // final M-step does not affect the output (only R is used)
            cap_nk<<<Bb * Kk, 256, 0, stream>>>(R, Nk);
            cap_mstep_gemm<<<dim3(Bb, 2, 2), 32, 0, stream>>>(R, x, S1, S2);
            cap_finalize<<<(Bb * Kk + 255) / 256, 256, 0, stream>>>(S1, S2, Nk, mu, is2, coef);
        }
    }

    cap_caps_gemm<<<dim3(Kk, NC), 64, 0, stream>>>(W, x, R, part);
    cap_reduce_squash<<<dim3(Kk, 2), 256, 0, stream>>>(part, out);
}